// MessagePassingBlock_30975304138980
// MI455X (gfx1250) — compile-verified
//
#include <hip/hip_runtime.h>
#include <hip/hip_bf16.h>

#define GDIM 512
#define LN_EPS 1e-5f

typedef __attribute__((ext_vector_type(2))) float v2f;
typedef __attribute__((ext_vector_type(8))) float v8f;

// ---------------------------------------------------------------------------
// Kernel 1: zero the workspace (agg [N*512] + deg [N])
// ---------------------------------------------------------------------------
__global__ void zero_ws_kernel(float* __restrict__ p, int n) {
    int i = blockIdx.x * blockDim.x + threadIdx.x;
    if (i < n) p[i] = 0.0f;
}

// ---------------------------------------------------------------------------
// Kernel 2: wave-per-edge scatter:  agg[src] += h[dst];  deg[src] += 1
// (the neighbor Linear is hoisted out of the edge loop algebraically)
// ---------------------------------------------------------------------------
__global__ __launch_bounds__(256) void scatter_kernel(
        const float* __restrict__ h, const int* __restrict__ ei,
        float* __restrict__ agg, float* __restrict__ deg, int E) {
    int wid  = (blockIdx.x * blockDim.x + threadIdx.x) >> 5;   // edge id
    int lane = threadIdx.x & 31;
    if (wid >= E) return;
    int s = ei[wid];         // edge_index[0][e]
    int d = ei[E + wid];     // edge_index[1][e]
    const float4* hp = (const float4*)(h + (size_t)d * GDIM);
    float* ap = agg + (size_t)s * GDIM;
#pragma unroll
    for (int i = 0; i < 4; ++i) {
        float4 v = hp[lane + 32 * i];
        int c = (lane + 32 * i) * 4;
        atomicAdd(ap + c + 0, v.x);
        atomicAdd(ap + c + 1, v.y);
        atomicAdd(ap + c + 2, v.z);
        atomicAdd(ap + c + 3, v.w);
    }
    if (lane == 0) atomicAdd(deg + s, 1.0f);
}

// ---------------------------------------------------------------------------
// Kernel 3: fused dual-GEMM (WMMA f32 16x16x4) + bias + ReLU + residual + LN
// Block = 256 threads = 8 waves; block computes 16 rows x 512 cols.
// Wave w handles N-tiles [64w, 64w+64). Both W_self and (agg/deg)@W_neigh
// accumulate into the same v8f accumulators.
// ---------------------------------------------------------------------------
__global__ __launch_bounds__(256, 2) void fused_gemm_ln_kernel(
        const float* __restrict__ h,
        const float* __restrict__ Ws,  const float* __restrict__ bsv,
        const float* __restrict__ Wn,  const float* __restrict__ bnv,
        const float* __restrict__ gamma, const float* __restrict__ beta,
        const float* __restrict__ agg, const float* __restrict__ deg,
        float* __restrict__ out) {
    __shared__ float xs[16][GDIM];

    const int tid  = threadIdx.x;
    const int lane = tid & 31;
    const int wave = tid >> 5;
    const int row0 = blockIdx.x * 16;

    const int m16 = lane & 15;           // row (A) / col (B) within a 16-tile
    const int kb  = (lane >> 4) * 2;     // K sub-offset within K=4 step
    const int row = row0 + m16;

    const float rd = 1.0f / fmaxf(deg[row], 1.0f);   // degree normalization

    const int n0 = wave * 64;

    v8f acc[4];
#pragma unroll
    for (int t = 0; t < 4; ++t) acc[t] = (v8f){0,0,0,0,0,0,0,0};

    const float* hrow = h   + (size_t)row * GDIM;
    const float* arow = agg + (size_t)row * GDIM;
    const float* wsrow[4];
    const float* wnrow[4];
#pragma unroll
    for (int t = 0; t < 4; ++t) {
        wsrow[t] = Ws + (size_t)(n0 + t * 16 + m16) * GDIM;  // B[k][n]=W[n][k]
        wnrow[t] = Wn + (size_t)(n0 + t * 16 + m16) * GDIM;
    }

#pragma unroll 2
    for (int k = 0; k < GDIM; k += 4) {
        v2f ah = *(const v2f*)(hrow + k + kb);
        v2f aa = *(const v2f*)(arow + k + kb);
        aa.x *= rd;
        aa.y *= rd;
#pragma unroll
        for (int t = 0; t < 4; ++t) {
            v2f bsf = *(const v2f*)(wsrow[t] + k + kb);
            acc[t] = __builtin_amdgcn_wmma_f32_16x16x4_f32(
                false, ah, false, bsf, (short)0, acc[t], false, false);
            v2f bnf = *(const v2f*)(wnrow[t] + k + kb);
            acc[t] = __builtin_amdgcn_wmma_f32_16x16x4_f32(
                false, aa, false, bnf, (short)0, acc[t], false, false);
        }
    }

    // Epilogue: bias + ReLU + residual, staged to LDS for the row-wise LN.
    // C/D layout: VGPR j -> lanes 0-15: M=j, N=lane; lanes 16-31: M=j+8.
    const int mhi = (lane >= 16) ? 8 : 0;
#pragma unroll
    for (int t = 0; t < 4; ++t) {
        int n = n0 + t * 16 + m16;
        float bias = bsv[n] + bnv[n];
#pragma unroll
        for (int j = 0; j < 8; ++j) {
            int m = mhi + j;
            float v = acc[t][j] + bias;
            v = fmaxf(v, 0.0f) + h[(size_t)(row0 + m) * GDIM + n];
            xs[m][n] = v;
        }
    }
    __syncthreads();

    // LayerNorm: 16 threads per row, each covers 32 strided columns,
    // reduce across the 16-lane half-wave with shfl_xor.
    const int r  = tid >> 4;     // row 0..15 (half-wave granularity)
    const int cg = tid & 15;
    float s = 0.0f, s2 = 0.0f;
#pragma unroll 8
    for (int i = 0; i < 32; ++i) {
        float v = xs[r][cg + 16 * i];
        s  += v;
        s2 += v * v;
    }
#pragma unroll
    for (int off = 8; off > 0; off >>= 1) {
        s  += __shfl_xor(s,  off, 16);
        s2 += __shfl_xor(s2, off, 16);
    }
    const float mean = s * (1.0f / GDIM);
    const float var  = s2 * (1.0f / GDIM) - mean * mean;
    const float rstd = rsqrtf(var + LN_EPS);
    float* orow = out + (size_t)(row0 + r) * GDIM;
#pragma unroll 8
    for (int i = 0; i < 32; ++i) {
        int c = cg + 16 * i;
        orow[c] = (xs[r][c] - mean) * rstd * gamma[c] + beta[c];
    }
}

// ---------------------------------------------------------------------------
extern "C" void kernel_launch(void* const* d_in, const int* in_sizes, int n_in,
                              void* d_out, int out_size, void* d_ws, size_t ws_size,
                              hipStream_t stream) {
    const float* h     = (const float*)d_in[0];
    const int*   ei    = (const int*)  d_in[1];
    const float* Ws    = (const float*)d_in[2];
    const float* bs    = (const float*)d_in[3];
    const float* Wn    = (const float*)d_in[4];
    const float* bn    = (const float*)d_in[5];
    const float* gamma = (const float*)d_in[6];
    const float* beta  = (const float*)d_in[7];
    float* out = (float*)d_out;

    const int N = in_sizes[0] / GDIM;    // 10000
    const int E = in_sizes[1] / 2;       // 160000

    float* agg = (float*)d_ws;                 // N*512 floats
    float* deg = agg + (size_t)N * GDIM;       // N floats

    const int ztotal = N * GDIM + N;
    zero_ws_kernel<<<(ztotal + 255) / 256, 256, 0, stream>>>(agg, ztotal);

    scatter_kernel<<<(E + 7) / 8, 256, 0, stream>>>(h, ei, agg, deg, E);

    fused_gemm_ln_kernel<<<N / 16, 256, 0, stream>>>(
        h, Ws, bs, Wn, bn, gamma, beta, agg, deg, out);
}